// SoftMinLayer_46892452938400
// MI455X (gfx1250) — compile-verified
//
#include <hip/hip_runtime.h>
#include <math.h>

typedef __attribute__((ext_vector_type(2))) float v2f;
typedef __attribute__((ext_vector_type(8))) float v8f;

#define SEG        5
#define SEGLEN     100            // L / SEG
#define WLEN       500
#define WSTEP      250
#define WIN_PER_WAVE  3           // 15 of 16 WMMA rows used
#define WAVES_PER_BLK 8
#define WIN_PER_BLK   (WIN_PER_WAVE * WAVES_PER_BLK)        // 24
#define IN_PER_BLK    (WSTEP * (WIN_PER_BLK - 1) + WLEN)    // 6250 floats
#define KSTEPS        (SEGLEN / 4)                          // 25 wmma f32 16x16x4 steps

__global__ void softmin_zero(float* out) { out[0] = 0.0f; }

__global__ __launch_bounds__(256) void softmin_dtw_wmma(
    const float* __restrict__ input, const float* __restrict__ S,
    float* __restrict__ out, int Q, int W, float invW)
{
    __shared__ float lds_in[IN_PER_BLK];                  // staged input slice
    __shared__ float lds_S[WLEN];                         // staged S
    __shared__ float lds_xy[WAVES_PER_BLK][15][SEG];      // D tile rows 0..14, cols 0..4
    __shared__ float lds_xx[WAVES_PER_BLK][15];
    __shared__ float lds_yy[WAVES_PER_BLK][SEG];
    __shared__ float lds_part[WAVES_PER_BLK];

    const int tid  = threadIdx.x;
    const int lane = tid & 31;
    const int wv   = tid >> 5;

    // ---- stage input (coalesced) + S into LDS ----
    const int gbase = blockIdx.x * (WSTEP * WIN_PER_BLK); // 6000 floats per block
    for (int k = tid; k < IN_PER_BLK; k += 256) {
        int g = gbase + k;
        lds_in[k] = (g < Q) ? input[g] : 0.0f;
    }
    for (int k = tid; k < WLEN; k += 256) lds_S[k] = S[k];
    __syncthreads();

    // ---- per-lane WMMA fragment addressing (f32 16x16x4 layout) ----
    // A 16x4: lanes 0-15 -> M=lane, K-half 0 (K=0,1 in v0,v1); lanes 16-31 -> K-half 1 (K=2,3)
    const int khalf = lane >> 4;
    int m = lane & 15; if (m > 14) m = 14;                // row 15 duplicates row 14 (ignored)
    const int wc = m / SEG;                               // window within wave tile (0..2)
    const int si = m - SEG * wc;                          // segment index i
    const int offA = wv * (WIN_PER_WAVE * WSTEP) + wc * WSTEP + si * SEGLEN + 2 * khalf;
    int jb = lane & 15; if (jb > SEG - 1) jb = SEG - 1;   // cols 5..15 duplicate col 4 (ignored)
    const int offB = jb * SEGLEN + 2 * khalf;

    // ---- K loop: 25 x V_WMMA_F32_16X16X4_F32, plus xx/yy partials ----
    v8f acc = {};
    float xxp = 0.0f, yyp = 0.0f;
#pragma unroll
    for (int t = 0; t < KSTEPS; ++t) {
        v2f a = *(const v2f*)&lds_in[offA + 4 * t];
        v2f b = *(const v2f*)&lds_S[offB + 4 * t];
        xxp += a.x * a.x + a.y * a.y;
        yyp += b.x * b.x + b.y * b.y;
        acc = __builtin_amdgcn_wmma_f32_16x16x4_f32(
            /*neg_a=*/false, a, /*neg_b=*/false, b,
            /*c_mod=*/(short)0, acc, /*reuse_a=*/false, /*reuse_b=*/false);
    }

    // combine the two K-halves (lanes l and l+16 hold complementary K slices)
    const float xx = xxp + __shfl_xor(xxp, 16);
    const float yy = yyp + __shfl_xor(yyp, 16);

    // ---- scatter results to LDS ----
    if (lane < 15) lds_xx[wv][lane] = xx;                 // lane<15: m==lane, unclamped
    if (lane < SEG) lds_yy[wv][lane] = yy;
    {
        const int j = lane & 15;
        if (j < SEG) {
#pragma unroll
            for (int p = 0; p < 8; ++p) {
                const int mr = p + 8 * khalf;             // D row: vgpr p, lanes16-31 -> +8
                if (mr < 15) lds_xy[wv][mr][j] = acc[p];
            }
        }
    }
    __syncthreads();

    // ---- banded DTW (radius 1) for 3 windows per wave, lanes 0..2 ----
    float contrib = 0.0f;
    if (lane < WIN_PER_WAVE) {
        const int w = blockIdx.x * WIN_PER_BLK + wv * WIN_PER_WAVE + lane;
        if (w < W) {
            float C[SEG][SEG];
#pragma unroll
            for (int i = 0; i < SEG; ++i)
#pragma unroll
                for (int j = 0; j < SEG; ++j)
                    C[i][j] = lds_xx[wv][SEG * lane + i] + lds_yy[wv][j]
                              - 2.0f * lds_xy[wv][SEG * lane + i][j];
            float cum[SEG + 1][SEG + 1];
#pragma unroll
            for (int i = 0; i <= SEG; ++i)
#pragma unroll
                for (int j = 0; j <= SEG; ++j)
                    cum[i][j] = INFINITY;
            cum[0][0] = 0.0f;
#pragma unroll
            for (int i = 0; i < SEG; ++i) {
                const int lo = (i - 1 < 0) ? 0 : i - 1;
                const int hi = (i + 1 > SEG - 1) ? SEG - 1 : i + 1;
#pragma unroll
                for (int j = 0; j < SEG; ++j) {
                    if (j < lo || j > hi) continue;
                    const float prev = fminf(fminf(cum[i][j + 1], cum[i + 1][j]), cum[i][j]);
                    cum[i + 1][j + 1] = C[i][j] + prev;
                }
            }
            const float d = sqrtf(cum[SEG][SEG]);
            contrib = expf(-0.01f * d) * invW;
        }
    }

    // ---- wave reduce, then block reduce, one atomic per block ----
#pragma unroll
    for (int off = 16; off >= 1; off >>= 1) contrib += __shfl_xor(contrib, off);
    if (lane == 0) lds_part[wv] = contrib;
    __syncthreads();
    if (tid == 0) {
        float s = 0.0f;
#pragma unroll
        for (int v = 0; v < WAVES_PER_BLK; ++v) s += lds_part[v];
        atomicAdd(out, s);
    }
}

extern "C" void kernel_launch(void* const* d_in, const int* in_sizes, int n_in,
                              void* d_out, int out_size, void* d_ws, size_t ws_size,
                              hipStream_t stream) {
    const float* input = (const float*)d_in[0];
    const float* S     = (const float*)d_in[1];
    float* out = (float*)d_out;

    const int Q    = in_sizes[0];
    const int L    = in_sizes[1];          // 500
    const int step = L / 2;                // 250
    const int W    = (Q - L + step - 1) / step;   // 79998
    const float invW = 1.0f / (float)W;
    const int blocks = (W + WIN_PER_BLK - 1) / WIN_PER_BLK;

    softmin_zero<<<1, 1, 0, stream>>>(out);
    softmin_dtw_wmma<<<blocks, 256, 0, stream>>>(input, S, out, Q, W, invW);
}